// OnlineTripletLoss_54133767799026
// MI455X (gfx1250) — compile-verified
//
#include <hip/hip_runtime.h>
#include <hip/hip_bf16.h>

#define N_SAMPLES 512
#define EMB_DIM   256
#define MARGIN    3.0f

typedef float v2f __attribute__((ext_vector_type(2)));
typedef float v8f __attribute__((ext_vector_type(8)));
typedef unsigned int v4u __attribute__((ext_vector_type(4)));
typedef int v8i __attribute__((ext_vector_type(8)));
typedef int v4i __attribute__((ext_vector_type(4)));

// ---------------------------------------------------------------------------
// Kernel 1: row squared norms  sq[i] = sum_k E[i][k]^2 ; also zero accumulators
// ---------------------------------------------------------------------------
__global__ __launch_bounds__(256)
void OTL_rownorm_kernel(const float* __restrict__ E,
                        float* __restrict__ sq,
                        float* __restrict__ loss_sum,
                        unsigned* __restrict__ trip_cnt) {
    __shared__ float red[256];
    const int i = blockIdx.x;
    const int t = threadIdx.x;
    float v = E[i * EMB_DIM + t];
    red[t] = v * v;
    __syncthreads();
#pragma unroll
    for (int s = 128; s > 0; s >>= 1) {
        if (t < s) red[t] += red[t + s];
        __syncthreads();
    }
    if (t == 0) {
        sq[i] = red[0];
        if (i == 0) { *loss_sum = 0.0f; *trip_cnt = 0u; }
    }
}

// ---------------------------------------------------------------------------
// Kernel 2: Gram matrix via V_WMMA_F32_16X16X4_F32, fused into distance matrix
//   Dmat[m][n] = sq[m] + sq[n] - 2 * dot(E[m], E[n])
// One wave (32 threads) computes one 16x16 tile; K loop = 256/4 = 64 WMMAs.
// Fragment layouts per CDNA5 ISA 7.12.2 (32-bit A 16x4 and mirrored B 4x16):
//   lanes 0-15 : VGPR0=K0, VGPR1=K1 ; lanes 16-31 : VGPR0=K2, VGPR1=K3
// For a Gram matrix B[k][n] = E[n][k], so A and B loads share one pattern.
// ---------------------------------------------------------------------------
__global__ __launch_bounds__(32)
void OTL_gram_wmma_kernel(const float* __restrict__ E,
                          const float* __restrict__ sq,
                          float* __restrict__ Dmat) {
    const int tm   = blockIdx.x;          // row tile
    const int tn   = blockIdx.y;          // col tile
    const int lane = threadIdx.x;         // 0..31 (wave32)
    const int half = lane >> 4;           // 0: K{0,1}, 1: K{2,3}
    const int l16  = lane & 15;

    const int rowA = tm * 16 + l16;       // M index held by this lane (A frag)
    const int rowB = tn * 16 + l16;       // N index held by this lane (B frag)

    v8f c = {};
#pragma unroll 4
    for (int k0 = 0; k0 < EMB_DIM; k0 += 4) {
        v2f a, b;
        const int ka = k0 + 2 * half;
        a.x = E[rowA * EMB_DIM + ka + 0];
        a.y = E[rowA * EMB_DIM + ka + 1];
        b.x = E[rowB * EMB_DIM + ka + 0];
        b.y = E[rowB * EMB_DIM + ka + 1];
        // 8 args: (neg_a, A, neg_b, B, c_mod, C, reuse_a, reuse_b)
        c = __builtin_amdgcn_wmma_f32_16x16x4_f32(
                false, a, false, b, (short)0, c, false, false);
    }

    // C/D layout: lanes 0-15 -> M = v (v=0..7), lanes 16-31 -> M = v+8; N = l16
    const float sqn = sq[tn * 16 + l16];
#pragma unroll
    for (int v = 0; v < 8; ++v) {
        const int m = tm * 16 + v + 8 * half;
        Dmat[m * N_SAMPLES + tn * 16 + l16] = sq[m] + sqn - 2.0f * c[v];
    }
}

// ---------------------------------------------------------------------------
// Kernel 3: per positive-pair (i,j) random-hard-negative argmax + hinge loss.
// Block = 8 waves sharing anchor row i. D[i,:] is staged into LDS by the
// Tensor Data Mover (TENSOR_LOAD_TO_LDS, tracked with TENSORcnt) issued by
// wave 0 while all threads stage labels; wave w handles j = blockIdx.y*8 + w.
// ---------------------------------------------------------------------------
__global__ __launch_bounds__(256)
void OTL_triplet_scan_kernel(const float* __restrict__ Dmat,
                             const int* __restrict__ labels,
                             const float* __restrict__ randu,
                             float* __restrict__ loss_sum,
                             unsigned* __restrict__ trip_cnt) {
    __shared__ float sD[N_SAMPLES];
    __shared__ int   sL[N_SAMPLES];

    const int i   = blockIdx.x;
    const int tid = threadIdx.y * 32 + threadIdx.x;

#if __has_builtin(__builtin_amdgcn_tensor_load_to_lds)
    if (threadIdx.y == 0) {
        // --- Tensor DMA descriptor (D#) for a 512x1 f32 tile, global -> LDS ---
        // Flat->LDS mapping truncates to addr[31:0] (ISA 10.2), so the low 32
        // bits of the generic pointer are the wave-relative LDS byte offset.
        const unsigned lds_off = (unsigned)(uintptr_t)(void*)sD;
        const unsigned long long ga =
            (unsigned long long)(uintptr_t)(Dmat + (size_t)i * N_SAMPLES);

        v4u g0;
        g0.x = 1u;                                   // count=1 (valid), user mode
        g0.y = lds_off;                              // lds_addr [63:32]
        g0.z = (unsigned)(ga & 0xFFFFFFFFu);         // global_addr[31:0]
        g0.w = (unsigned)((ga >> 32) & 0x01FFFFFFu)  // global_addr[56:32]
             | (2u << 30);                           // type=2 ("image")

        v8i g1;
        g1[0] = (int)(2u << 16);                     // wg_mask=0, data_size=2 (4B)
        g1[1] = (int)((unsigned)N_SAMPLES << 16);    // tensor_dim0[15:0]=512
        g1[2] = (int)(1u << 16);                     // tensor_dim0[31:16]=0, tensor_dim1=1
        g1[3] = (int)((unsigned)N_SAMPLES << 16);    // tile_dim0=512
        g1[4] = 1;                                   // tile_dim1=1, tile_dim2=0
        g1[5] = N_SAMPLES;                           // tensor_dim0_stride[31:0]=512
        g1[6] = (int)((unsigned)N_SAMPLES << 16);    // stride[47:32]=0, dim1_stride[15:0]
        g1[7] = 0;                                   // dim1_stride[47:16]=0

        const v4i gz4 = {0, 0, 0, 0};                // groups 2/3 unused (<=2D)
        const v8i gz8 = {0, 0, 0, 0, 0, 0, 0, 0};    // extra group (clang-23 form)
        // 6-arg form: (g0, g1, g2, g3, g4, cpol)
        __builtin_amdgcn_tensor_load_to_lds(g0, g1, gz4, gz4, gz8, 0);
        __builtin_amdgcn_s_wait_tensorcnt(0);        // TDM done before barrier
    }
#else
    for (int t = tid; t < N_SAMPLES; t += 256) sD[t] = Dmat[i * N_SAMPLES + t];
#endif
    for (int t = tid; t < N_SAMPLES; t += 256) sL[t] = labels[t];
    __syncthreads();

    const int j  = blockIdx.y * 8 + threadIdx.y;
    const int li = sL[i];
    if (j <= i || sL[j] != li) return;       // wave-uniform exit

    const float Dij  = sD[j];
    const int   lane = threadIdx.x;
    const float* __restrict__ rrow =
        randu + ((size_t)i * N_SAMPLES + j) * N_SAMPLES;
    __builtin_prefetch(rrow + lane * 16, 0, 0);   // global_prefetch_b8

    float best = -2.0f;   // below the -1.0 "invalid" sentinel
    int   bidx = 0;
    int   anyv = 0;
#pragma unroll 4
    for (int n = lane; n < N_SAMPLES; n += 32) {
        const bool valid = (sL[n] != li) && (Dij - sD[n] + MARGIN > 0.0f);
        const float sc = valid ? rrow[n] : -1.0f;
        anyv |= (int)valid;
        if (sc > best) { best = sc; bidx = n; }   // strict > keeps lowest n
    }

    // wave32 argmax reduction, first-index tie-break (matches jnp.argmax)
#pragma unroll
    for (int off = 16; off > 0; off >>= 1) {
        const float ob = __shfl_xor(best, off, 32);
        const int   oi = __shfl_xor(bidx, off, 32);
        if (ob > best || (ob == best && oi < bidx)) { best = ob; bidx = oi; }
    }
    anyv = __any(anyv);

    if (lane == 0 && anyv) {
        // chosen negative is valid => hinge is strictly positive => relu = L
        const float loss = Dij - sD[bidx] + MARGIN;
        atomicAdd(loss_sum, loss);
        atomicAdd(trip_cnt, 1u);
    }
}

// ---------------------------------------------------------------------------
// Kernel 4: finalize  loss = sum / max(count, 1);  out[1] = n_triplets
// ---------------------------------------------------------------------------
__global__ void OTL_finalize_kernel(const float* __restrict__ loss_sum,
                                    const unsigned* __restrict__ trip_cnt,
                                    float* __restrict__ out) {
    if (threadIdx.x == 0 && blockIdx.x == 0) {
        const unsigned c = *trip_cnt;
        const float denom = (float)(c > 0u ? c : 1u);
        out[0] = (*loss_sum) / denom;
        out[1] = (float)c;
    }
}

extern "C" void kernel_launch(void* const* d_in, const int* in_sizes, int n_in,
                              void* d_out, int out_size, void* d_ws, size_t ws_size,
                              hipStream_t stream) {
    (void)in_sizes; (void)n_in; (void)out_size; (void)ws_size;

    const float* E      = (const float*)d_in[0];   // [512, 256] f32
    const int*   labels = (const int*)d_in[1];     // [512] i32
    const float* randu  = (const float*)d_in[2];   // [512, 512, 512] f32

    char* ws = (char*)d_ws;
    float*    Dmat     = (float*)ws;                                   // 1 MB
    float*    sq       = (float*)(ws + (size_t)N_SAMPLES * N_SAMPLES * 4);
    float*    loss_sum = (float*)(ws + (size_t)N_SAMPLES * N_SAMPLES * 4
                                     + (size_t)N_SAMPLES * 4);
    unsigned* trip_cnt = (unsigned*)(loss_sum + 1);

    OTL_rownorm_kernel<<<N_SAMPLES, 256, 0, stream>>>(E, sq, loss_sum, trip_cnt);

    OTL_gram_wmma_kernel<<<dim3(N_SAMPLES / 16, N_SAMPLES / 16), 32, 0, stream>>>(
        E, sq, Dmat);

    OTL_triplet_scan_kernel<<<dim3(N_SAMPLES, N_SAMPLES / 8), dim3(32, 8), 0, stream>>>(
        Dmat, labels, randu, loss_sum, trip_cnt);

    OTL_finalize_kernel<<<1, 32, 0, stream>>>(loss_sum, trip_cnt, (float*)d_out);
}